// DynamicSpatialBlock_22711787061803
// MI455X (gfx1250) — compile-verified
//
#include <hip/hip_runtime.h>
#include <hip/hip_bf16.h>
#include <math.h>

typedef __attribute__((ext_vector_type(2))) float v2f;
typedef __attribute__((ext_vector_type(8))) float v8f;

#define B_    256
#define NIN   64
#define T_    2000
#define INP   2080          // 64*65/2
#define HID   520
#define OUTF  4160          // 64*64 + 64
#define SOFT_T 0.1f

static __device__ __forceinline__ v8f wmma_f32(v2f a, v2f b, v8f c) {
  return __builtin_amdgcn_wmma_f32_16x16x4_f32(false, a, false, b, (short)0, c, false, false);
}

// ======= Kernel 1: Gram via f32 WMMA, fused mean, cov = (G - T mu muT)/(T-1) =======
// 512 threads = 16 waves; wave (mi,ni) owns one 16x16 tile of the 64x64 output.
// Double-buffered LDS (pad 68 for b128 alignment); steady-state prefetch is two
// unconditional global_load_b128 per thread; only the tail chunk is guarded.
__global__ void __launch_bounds__(512) k_cov(const float* __restrict__ x,
                                             float* __restrict__ mean,
                                             float* __restrict__ cov) {
  __shared__ float lx[2][64][68];
  __shared__ float smean[64];
  int b = blockIdx.x;
  const float* xb = x + (size_t)b * NIN * T_;
  int tid = threadIdx.x;
  int lane = tid & 31, wid = tid >> 5;
  int mi = wid >> 2, ni = wid & 3;
  int hi = lane >> 4, lo = lane & 15;
  int ca = mi * 16 + lo, cb = ni * 16 + lo;

  int lrow = tid >> 3;             // 0..63 : channel row this thread loads
  int lcol = (tid & 7) * 8;        // 0,8,..,56 : 8 contiguous t per thread
  const float* xrow = xb + (size_t)lrow * T_ + lcol;

  // prologue: chunk 0 (fully in range) via two b128 loads
  {
    float4 p0 = *(const float4*)(xrow + 0);
    float4 p1 = *(const float4*)(xrow + 4);
    *(float4*)&lx[0][lrow][lcol]     = p0;
    *(float4*)&lx[0][lrow][lcol + 4] = p1;
  }
  __syncthreads();

  float rowsum = 0.f;
  v8f acc = {};
  const int NCH = (T_ + 63) / 64;  // 32
  for (int ch = 0; ch < NCH; ++ch) {
    int cur = ch & 1, nxt = cur ^ 1;
    bool more = (ch + 1 < NCH);
    float4 s0, s1;
    if (more) {                    // prefetch next chunk into registers
      int k0n = (ch + 1) * 64;
      if (k0n + 64 <= T_) {        // full chunk: unconditional 128-bit loads
        s0 = *(const float4*)(xrow + k0n);
        s1 = *(const float4*)(xrow + k0n + 4);
      } else {                     // tail chunk: per-element guard + zero pad
        float tmp[8];
        for (int s = 0; s < 8; ++s) {
          int t = k0n + lcol + s;
          tmp[s] = (t < T_) ? xb[(size_t)lrow * T_ + t] : 0.f;
        }
        s0 = make_float4(tmp[0], tmp[1], tmp[2], tmp[3]);
        s1 = make_float4(tmp[4], tmp[5], tmp[6], tmp[7]);
      }
    }
    // 16 WMMA K-steps on current chunk (hides the prefetch latency)
    for (int kk = 0; kk < 64; kk += 4) {
      int ko = kk + 2 * hi;
      v2f a, bf;
      a.x  = lx[cur][ca][ko]; a.y  = lx[cur][ca][ko + 1];
      bf.x = lx[cur][cb][ko]; bf.y = lx[cur][cb][ko + 1];
      acc = wmma_f32(a, bf, acc);
    }
    if (tid < 64) {                // waves 0-1: per-channel sum of this chunk
      for (int kk = 0; kk < 64; ++kk) rowsum += lx[cur][tid][kk];
    }
    if (more) {
      *(float4*)&lx[nxt][lrow][lcol]     = s0;
      *(float4*)&lx[nxt][lrow][lcol + 4] = s1;
    }
    __syncthreads();               // one barrier per chunk
  }

  if (tid < 64) {
    float m = rowsum * (1.0f / T_);
    smean[tid] = m;
    mean[b * 64 + tid] = m;        // publish for k_wb
  }
  __syncthreads();

  int col = ni * 16 + lo;
  float mu_c = smean[col];
  const float inv = 1.0f / (float)(T_ - 1);
  for (int r = 0; r < 8; ++r) {
    int row = mi * 16 + r + 8 * hi;
    cov[(size_t)b * 4096 + row * 64 + col] =
        (acc[r] - (float)T_ * smean[row] * mu_c) * inv;
  }
}

// ======= Kernel 2: parallel tournament Jacobi + matrix-log -> upper-tri feats =======
__global__ void __launch_bounds__(256) k_eig(const float* __restrict__ cov,
                                             float* __restrict__ rstd,
                                             float* __restrict__ feats) {
  __shared__ float A[64][65];
  __shared__ float V[64][65];
  __shared__ float le[64];
  __shared__ float cc[32], ss[32];
  __shared__ int   pl[32], ql[32];
  int b = blockIdx.x, tid = threadIdx.x;
  const float* cbp = cov + (size_t)b * 4096;
  for (int i = tid; i < 4096; i += 256) {
    int r = i >> 6, c = i & 63;
    A[r][c] = cbp[i];
    V[r][c] = (r == c) ? 1.f : 0.f;
  }
  __syncthreads();
  if (tid < 64) rstd[b * 64 + tid] = rsqrtf(fmaxf(A[tid][tid], 1e-20f));

  int jcol = tid & 63, grp = tid >> 6;          // 4 groups of 8 pairs each
  for (int sweep = 0; sweep < 10; ++sweep) {
    for (int r = 0; r < 63; ++r) {
      if (tid < 32) {                           // round-robin pairing (slot 0 fixed)
        int sp = tid, sq = 63 - tid;
        int p = (sp == 0) ? 0 : 1 + (sp - 1 + r) % 63;
        int q = 1 + (sq - 1 + r) % 63;
        float apq = A[p][q], app = A[p][p], aqq = A[q][q];
        float c = 1.f, s = 0.f;
        if (fabsf(apq) > 1e-12f) {
          float th = 0.5f * (aqq - app) / apq;
          float t = 1.f / (fabsf(th) + sqrtf(th * th + 1.f));
          if (th < 0.f) t = -t;
          c = rsqrtf(t * t + 1.f);
          s = t * c;
        }
        pl[tid] = p; ql[tid] = q; cc[tid] = c; ss[tid] = s;
      }
      __syncthreads();
      for (int i = grp * 8; i < grp * 8 + 8; ++i) {   // rows: A <- J^T A
        int p = pl[i], q = ql[i];
        float c = cc[i], s = ss[i];
        float apj = A[p][jcol], aqj = A[q][jcol];
        A[p][jcol] = c * apj - s * aqj;
        A[q][jcol] = s * apj + c * aqj;
      }
      __syncthreads();
      for (int i = grp * 8; i < grp * 8 + 8; ++i) {   // cols: A <- A J, V <- V J
        int p = pl[i], q = ql[i];
        float c = cc[i], s = ss[i];
        float ajp = A[jcol][p], ajq = A[jcol][q];
        A[jcol][p] = c * ajp - s * ajq;
        A[jcol][q] = s * ajp + c * ajq;
        float vjp = V[jcol][p], vjq = V[jcol][q];
        V[jcol][p] = c * vjp - s * vjq;
        V[jcol][q] = s * vjp + c * vjq;
      }
      __syncthreads();
    }
  }

  if (tid < 64) le[tid] = logf(fmaxf(A[tid][tid], 1e-10f));
  __syncthreads();
  float* fb = feats + (size_t)b * INP;
  for (int n = tid; n < INP; n += 256) {        // upper-tri linear index -> (i,j)
    int i = 0, base = 0;
    while (base + (64 - i) <= n) { base += 64 - i; ++i; }
    int j = i + (n - base);
    float sum = 0.f;
    for (int k = 0; k < 64; ++k) sum += V[i][k] * le[k] * V[j][k];
    fb[n] = sum;                                // log_cov[i][j], i<=j
  }
}

// ======= Kernel 3: h = relu(feats @ W1^T + b1) via WMMA (M=256,N=520,K=2080) =======
__global__ void __launch_bounds__(256) k_mlp1(const float* __restrict__ feats,
                                              const float* __restrict__ W1,
                                              const float* __restrict__ b1,
                                              float* __restrict__ h) {
  int lane = threadIdx.x & 31, wid = threadIdx.x >> 5;
  int wl = blockIdx.x * 8 + wid;              // 66 blocks * 8 waves = 528 = 16*33
  const int NT = 33;
  int mi = wl / NT, ni = wl % NT;
  int hi = lane >> 4, lo = lane & 15;
  int m = mi * 16 + lo;
  int n = ni * 16 + lo;
  bool nok = (n < HID);
  const float* arow = feats + (size_t)m * INP;
  const float* brow = W1 + (size_t)(nok ? n : 0) * INP;
  v8f acc = {};
  for (int k = 0; k < INP; k += 4) {          // 520 exact K-steps
    int ko = k + 2 * hi;
    v2f a, bf;
    a.x = arow[ko]; a.y = arow[ko + 1];
    if (nok) { bf.x = brow[ko]; bf.y = brow[ko + 1]; }
    else     { bf.x = 0.f;      bf.y = 0.f; }
    acc = wmma_f32(a, bf, acc);
  }
  if (nok) {
    float bias = b1[n];
    for (int r = 0; r < 8; ++r) {
      int row = mi * 16 + r + 8 * hi;
      h[(size_t)row * HID + n] = fmaxf(acc[r] + bias, 0.f);
    }
  }
}

// ======= Kernel 4: feats2 = h @ W2^T + b2 + soft-threshold (M=256,N=4160,K=520) =======
__global__ void __launch_bounds__(256) k_mlp2(const float* __restrict__ h,
                                              const float* __restrict__ W2,
                                              const float* __restrict__ b2,
                                              float* __restrict__ feats2) {
  int lane = threadIdx.x & 31, wid = threadIdx.x >> 5;
  int wl = blockIdx.x * 8 + wid;              // 520 blocks * 8 = 4160 = 16*260
  const int NT = 260;
  int mi = wl / NT, ni = wl % NT;
  int hi = lane >> 4, lo = lane & 15;
  const float* arow = h  + (size_t)(mi * 16 + lo) * HID;
  int n = ni * 16 + lo;
  const float* brow = W2 + (size_t)n * HID;
  v8f acc = {};
  for (int k = 0; k < HID; k += 4) {
    int ko = k + 2 * hi;
    v2f a, bf;
    a.x  = arow[ko]; a.y  = arow[ko + 1];
    bf.x = brow[ko]; bf.y = brow[ko + 1];
    acc = wmma_f32(a, bf, acc);
  }
  float bias = b2[n];
  bool is_w = (n < 4096);
  for (int r = 0; r < 8; ++r) {
    int row = mi * 16 + r + 8 * hi;
    float v = acc[r] + bias;
    if (is_w) v = fmaxf(v - SOFT_T, 0.f) + fminf(v + SOFT_T, 0.f);
    feats2[(size_t)row * OUTF + n] = v;
  }
}

// ======= Kernel 5: fold normalization: weff = w*rstd, beff = b - weff@mu =======
__global__ void __launch_bounds__(64) k_wb(const float* __restrict__ feats2,
                                           const float* __restrict__ rstd,
                                           const float* __restrict__ mean,
                                           float* __restrict__ weff,
                                           float* __restrict__ beff) {
  int b = blockIdx.x, o = threadIdx.x;
  const float* f = feats2 + (size_t)b * OUTF;
  float accv = 0.f;
  for (int c = 0; c < 64; ++c) {
    float wv = f[o * 64 + c] * rstd[b * 64 + c];
    weff[(size_t)b * 4096 + o * 64 + c] = wv;
    accv += wv * mean[b * 64 + c];
  }
  beff[b * 64 + o] = f[4096 + o] - accv;
}

// ======= Kernel 6: out[b] = weff[b] @ x[b] + beff[b]  (M=64, N=2000, K=64) =======
// One wave per 16-wide N-tile; 4 accumulators cover all 64 output rows so each
// x B-frag is loaded ONCE (x read exactly once from HBM). weff/beff staged in LDS.
__global__ void __launch_bounds__(256) k_out(const float* __restrict__ x,
                                             const float* __restrict__ weff,
                                             const float* __restrict__ beff,
                                             float* __restrict__ out) {
  __shared__ float lw[64][68];
  __shared__ float lb[64];
  int b = blockIdx.y;
  int tid = threadIdx.x;
  // stage weff[b] (64x64) and beff[b] in LDS, b128 loads/stores
  {
    int row = tid >> 2, colb = (tid & 3) * 16;
    const float* src = weff + (size_t)b * 4096 + (size_t)row * 64 + colb;
    for (int s = 0; s < 16; s += 4)
      *(float4*)&lw[row][colb + s] = *(const float4*)(src + s);
    if (tid < 64) lb[tid] = beff[b * 64 + tid];
  }
  __syncthreads();

  int lane = tid & 31, wid = tid >> 5;
  int ni = blockIdx.x * 8 + wid;
  if (ni >= 125) return;                     // 125 N-tiles of 16 (2000 exact)
  int hi = lane >> 4, lo = lane & 15;
  const float* xb = x + (size_t)b * NIN * T_;
  int n = ni * 16 + lo;

  v8f acc0 = {}, acc1 = {}, acc2 = {}, acc3 = {};
  for (int k = 0; k < 64; k += 4) {
    int ko = k + 2 * hi;
    v2f bf;
    bf.x = xb[(size_t)ko * T_ + n];
    bf.y = xb[(size_t)(ko + 1) * T_ + n];
    v2f a0, a1, a2, a3;
    a0.x = lw[ 0 + lo][ko]; a0.y = lw[ 0 + lo][ko + 1];
    a1.x = lw[16 + lo][ko]; a1.y = lw[16 + lo][ko + 1];
    a2.x = lw[32 + lo][ko]; a2.y = lw[32 + lo][ko + 1];
    a3.x = lw[48 + lo][ko]; a3.y = lw[48 + lo][ko + 1];
    acc0 = wmma_f32(a0, bf, acc0);
    acc1 = wmma_f32(a1, bf, acc1);
    acc2 = wmma_f32(a2, bf, acc2);
    acc3 = wmma_f32(a3, bf, acc3);
  }
  float* ob = out + (size_t)b * NIN * T_;
  for (int r = 0; r < 8; ++r) {
    int row = r + 8 * hi;
    ob[(size_t)(row     ) * T_ + n] = acc0[r] + lb[row     ];
    ob[(size_t)(row + 16) * T_ + n] = acc1[r] + lb[row + 16];
    ob[(size_t)(row + 32) * T_ + n] = acc2[r] + lb[row + 32];
    ob[(size_t)(row + 48) * T_ + n] = acc3[r] + lb[row + 48];
  }
}

// ======================= host launcher =======================
extern "C" void kernel_launch(void* const* d_in, const int* in_sizes, int n_in,
                              void* d_out, int out_size, void* d_ws, size_t ws_size,
                              hipStream_t stream) {
  const float* x  = (const float*)d_in[0];
  const float* W1 = (const float*)d_in[1];
  const float* b1 = (const float*)d_in[2];
  const float* W2 = (const float*)d_in[3];
  const float* b2 = (const float*)d_in[4];
  float* out = (float*)d_out;
  float* ws = (float*)d_ws;

  // workspace layout (floats)
  float* mean   = ws;                         // 16384
  float* rstd   = mean   + 16384;             // 16384
  float* cov    = rstd   + 16384;             // 256*4096
  float* feats  = cov    + 256 * 4096;        // 256*2080
  float* h      = feats  + 256 * INP;         // 256*520
  float* feats2 = h      + 256 * HID;         // 256*4160
  float* weff   = feats2 + 256 * OUTF;        // 256*4096
  float* beff   = weff   + 256 * 4096;        // 256*64

  k_cov <<<B_, 512, 0, stream>>>(x, mean, cov);          // fused mean + Gram
  k_eig <<<B_, 256, 0, stream>>>(cov, rstd, feats);      // fused rstd + Jacobi + log
  k_mlp1<<<66, 256, 0, stream>>>(feats, W1, b1, h);
  k_mlp2<<<520, 256, 0, stream>>>(h, W2, b2, feats2);
  k_wb  <<<B_, 64, 0, stream>>>(feats2, rstd, mean, weff, beff);
  dim3 g(16, B_);
  k_out <<<g, 256, 0, stream>>>(x, weff, beff, out);
}